// xModalKD_4209067950458
// MI455X (gfx1250) — compile-verified
//
#include <hip/hip_runtime.h>
#include <hip/hip_bf16.h>
#include <stdint.h>

#define LXM 0.05f
#define LSEG2D 1.0f
#define BN_EPS 1e-5f

typedef __attribute__((ext_vector_type(2))) float v2f;
typedef __attribute__((ext_vector_type(8))) float v8f;

#define ACT_NONE 0
#define ACT_RELU 1
#define ACT_SIG  2

// ------------------------------------------------------------------
// f32 WMMA GEMM: D[M,N] = act(A[M,K] @ B[K,N] + bias), one wave per 16x16 tile
// Uses V_WMMA_F32_16X16X4_F32 (CDNA5). M must be a multiple of 16 here.
// Out-of-range columns are handled branchlessly (clamped load + select) so the
// inner loop has no EXEC-mask manipulation around the WMMA.
// ------------------------------------------------------------------
__global__ void gemm_wmma_f32(const float* __restrict__ A, int lda,
                              const float* __restrict__ Bm, int ldb,
                              const float* __restrict__ bias,
                              float* __restrict__ D, int ldd,
                              int Mr, int Nc, int K, int act)
{
  const int tilesN = (Nc + 15) >> 4;
  const int wave   = (blockIdx.x * blockDim.x + threadIdx.x) >> 5;
  const int lane   = threadIdx.x & 31;
  const int tileRow = wave / tilesN;
  const int tileCol = wave - tileRow * tilesN;
  if (tileRow * 16 >= Mr) return;            // wave-uniform exit
  const int half = lane >> 4;                // lane group 0/1
  const int l16  = lane & 15;
  const int row  = tileRow * 16 + l16;       // A row for this lane
  const int col  = tileCol * 16 + l16;       // B/D column for this lane
  const bool colOK = (col < Nc);
  const float* Arow = A + (size_t)row * lda + 2 * half;
  const float* Bcol = Bm + (colOK ? col : (Nc - 1));   // clamped, always in-bounds
  v8f acc = {};
  for (int k0 = 0; k0 < K; k0 += 4) {
    const int ka = k0 + 2 * half;            // K pair handled by this lane-half
    v2f a, b;
    a.x = Arow[k0];
    a.y = Arow[k0 + 1];
    float bx = Bcol[(size_t)ka * ldb];
    float by = Bcol[(size_t)(ka + 1) * ldb];
    b.x = colOK ? bx : 0.f;                  // v_cndmask, no divergent branch
    b.y = colOK ? by : 0.f;
    acc = __builtin_amdgcn_wmma_f32_16x16x4_f32(
        /*neg_a=*/false, a, /*neg_b=*/false, b,
        /*c_mod=*/(short)0, acc, /*reuse_a=*/false, /*reuse_b=*/false);
  }
  if (colOK) {
    const float bv = bias ? bias[col] : 0.f;
    const int rbase = tileRow * 16 + 8 * half;
#pragma unroll
    for (int r = 0; r < 8; ++r) {
      const int rr = rbase + r;
      if (rr < Mr) {
        float v = acc[r] + bv;
        if (act == ACT_RELU)      v = fmaxf(v, 0.f);
        else if (act == ACT_SIG)  v = 1.f / (1.f + expf(-v));
        D[(size_t)rr * ldd + col] = v;
      }
    }
  }
}

static inline void launch_gemm(const float* A, int lda, const float* B, int ldb,
                               const float* bias, float* D, int ldd,
                               int M, int N, int K, int act, hipStream_t s)
{
  const int tilesM = (M + 15) / 16, tilesN = (N + 15) / 16;
  const long long waves = (long long)tilesM * tilesN;
  const int thr = 256;
  const unsigned blocks = (unsigned)((waves * 32 + thr - 1) / thr);
  gemm_wmma_f32<<<blocks, thr, 0, s>>>(A, lda, B, ldb, bias, D, ldd, M, N, K, act);
}

// ------------------------------------------------------------------ utilities
__global__ void zero_f32_k(float* p, int n) {
  int i = blockIdx.x * blockDim.x + threadIdx.x; if (i < n) p[i] = 0.f;
}
__global__ void zero_i32_k(int* p, int n) {
  int i = blockIdx.x * blockDim.x + threadIdx.x; if (i < n) p[i] = 0;
}

__global__ void scatter_counts_k(const int* __restrict__ coors, const int* __restrict__ lab,
                                 int* __restrict__ counts, int n, int C) {
  int i = blockIdx.x * blockDim.x + threadIdx.x;
  if (i < n) atomicAdd(&counts[(size_t)coors[i] * C + lab[i]], 1);
}

__global__ void vox_argmax_k(const int* __restrict__ counts, int* __restrict__ voxlab,
                             int nv, int C) {
  int v = blockIdx.x * blockDim.x + threadIdx.x;
  if (v >= nv) return;
  const int* row = counts + (size_t)v * C;
  int best = row[0], bi = 0;
  for (int c = 1; c < C; ++c) { int x = row[c]; if (x > best) { best = x; bi = c; } }
  voxlab[v] = bi;
}

__global__ void build_idx_k(const int* __restrict__ coors, const int* __restrict__ p2img,
                            int* __restrict__ idx, int nimg, int M, int Np) {
  int j = blockIdx.x * blockDim.x + threadIdx.x;
  if (j >= nimg) return;
  int b = j / M;
  idx[j] = coors[(size_t)b * Np + p2img[j]];
}

__global__ void gather_rows_k(const float* __restrict__ src, const int* __restrict__ idx,
                              float* __restrict__ dst, int rows, int cols) {
  int t = blockIdx.x * blockDim.x + threadIdx.x;
  if (t >= rows * cols) return;
  int r = t / cols, c = t - r * cols;
  dst[t] = src[(size_t)idx[r] * cols + c];
}

// deterministic column stats: one block per channel
__global__ void bn_stats_k(const float* __restrict__ y, int rows, int cols,
                           float* __restrict__ mu, float* __restrict__ var) {
  __shared__ float s1[256], s2[256];
  const int c = blockIdx.x, tid = threadIdx.x;
  float a = 0.f, b = 0.f;
  for (int r = tid; r < rows; r += blockDim.x) {
    float v = y[(size_t)r * cols + c]; a += v; b += v * v;
  }
  s1[tid] = a; s2[tid] = b; __syncthreads();
  for (int s = 128; s > 0; s >>= 1) {
    if (tid < s) { s1[tid] += s1[tid + s]; s2[tid] += s2[tid + s]; }
    __syncthreads();
  }
  if (tid == 0) { float m = s1[0] / rows; mu[c] = m; var[c] = s2[0] / rows - m * m; }
}

__global__ void fuse_k(const float* __restrict__ y1, const float* __restrict__ y2,
                       const float* __restrict__ mu1, const float* __restrict__ var1,
                       const float* __restrict__ g1,  const float* __restrict__ b1,
                       const float* __restrict__ mu2, const float* __restrict__ var2,
                       const float* __restrict__ g2,  const float* __restrict__ b2,
                       const float* __restrict__ w,   float* __restrict__ feats,
                       int rows, int Hc, int ldf, int off) {
  int t = blockIdx.x * blockDim.x + threadIdx.x;
  if (t >= rows * Hc) return;
  int r = t / Hc, c = t - r * Hc;
  float a = fmaxf((y1[t] - mu1[c]) * rsqrtf(var1[c] + BN_EPS) * g1[c] + b1[c], 0.f);
  float bb = fmaxf((y2[t] - mu2[c]) * rsqrtf(var2[c] + BN_EPS) * g2[c] + b2[c], 0.f);
  feats[(size_t)r * ldf + off + c] = a * w[r * 2] + bb * w[r * 2 + 1];
}

// ------------------------------------------------------------------ losses
__global__ void ce_partial_k(const float* __restrict__ logits, const int* __restrict__ lab,
                             int n, int C, float* __restrict__ partial) {
  __shared__ float sh[256];
  float acc = 0.f;
  for (int r = blockIdx.x * blockDim.x + threadIdx.x; r < n; r += gridDim.x * blockDim.x) {
    const float* L = logits + (size_t)r * C;
    float mx = L[0];
    for (int c = 1; c < C; ++c) mx = fmaxf(mx, L[c]);
    float se = 0.f;
    for (int c = 0; c < C; ++c) se += expf(L[c] - mx);
    acc += (logf(se) + mx - L[lab[r]]);
  }
  sh[threadIdx.x] = acc; __syncthreads();
  for (int s = 128; s > 0; s >>= 1) { if (threadIdx.x < s) sh[threadIdx.x] += sh[threadIdx.x + s]; __syncthreads(); }
  if (threadIdx.x == 0) partial[blockIdx.x] = sh[0];
}

__global__ void kl_partial_k(const float* __restrict__ fpred, const float* __restrict__ ppred,
                             int n, int C, float* __restrict__ partial) {
  __shared__ float sh[256];
  float acc = 0.f;
  for (int r = blockIdx.x * blockDim.x + threadIdx.x; r < n; r += gridDim.x * blockDim.x) {
    const float* F = fpred + (size_t)r * C;
    const float* P = ppred + (size_t)r * C;
    float mf = F[0], mp = P[0];
    for (int c = 1; c < C; ++c) { mf = fmaxf(mf, F[c]); mp = fmaxf(mp, P[c]); }
    float sf = 0.f, sp = 0.f;
    for (int c = 0; c < C; ++c) { sf += expf(F[c] - mf); sp += expf(P[c] - mp); }
    float lsf = logf(sf), lsp = logf(sp);
    for (int c = 0; c < C; ++c) {
      float lpf = F[c] - mf - lsf;
      float pf  = expf(lpf);
      float lsq = P[c] - mp - lsp;            // log_softmax(pts_pred)
      acc += pf * (lpf - lsq);
    }
  }
  sh[threadIdx.x] = acc; __syncthreads();
  for (int s = 128; s > 0; s >>= 1) { if (threadIdx.x < s) sh[threadIdx.x] += sh[threadIdx.x + s]; __syncthreads(); }
  if (threadIdx.x == 0) partial[blockIdx.x] = sh[0];
}

__global__ void finalize_scaled_k(const float* __restrict__ partial, int g, float scale,
                                  float* __restrict__ out) {
  __shared__ float sh[256];
  float a = (threadIdx.x < g) ? partial[threadIdx.x] : 0.f;
  sh[threadIdx.x] = a; __syncthreads();
  for (int s = 128; s > 0; s >>= 1) { if (threadIdx.x < s) sh[threadIdx.x] += sh[threadIdx.x + s]; __syncthreads(); }
  if (threadIdx.x == 0) out[0] += scale * sh[0];
}

// keys: [C][Npad] u64 = (orderedFloat(|fg-prob|) << 32) | fg ; padded rows -> 0
__global__ void build_keys_k(const float* __restrict__ logits, const int* __restrict__ lab,
                             int n, int Npad, int C, unsigned long long* __restrict__ keys) {
  int r = blockIdx.x * blockDim.x + threadIdx.x;
  if (r >= Npad) return;
  if (r >= n) { for (int c = 0; c < C; ++c) keys[(size_t)c * Npad + r] = 0ULL; return; }
  const float* L = logits + (size_t)r * C;
  float mx = L[0];
  for (int c = 1; c < C; ++c) mx = fmaxf(mx, L[c]);
  float se = 0.f;
  for (int c = 0; c < C; ++c) se += expf(L[c] - mx);
  const int lb = lab[r];
  for (int c = 0; c < C; ++c) {
    float prob = expf(L[c] - mx) / se;
    float fg = (c == lb) ? 1.f : 0.f;
    float err = fabsf(fg - prob);
    unsigned u = __float_as_uint(err);
    unsigned ord = (u & 0x80000000u) ? ~u : (u | 0x80000000u);
    keys[(size_t)c * Npad + r] = ((unsigned long long)ord << 32) | (unsigned long long)(c == lb);
  }
}

// descending bitonic sort pass over C independent arrays of length Npad (pow2)
__global__ void bitonic_pass_k(unsigned long long* __restrict__ keys, int Npad, int C,
                               int k, int j) {
  long long t = (long long)blockIdx.x * blockDim.x + threadIdx.x;
  if (t >= (long long)C * Npad) return;
  int c = (int)(t / Npad);
  int i = (int)(t - (long long)c * Npad);
  int l = i ^ j;
  if (l > i) {
    unsigned long long* base = keys + (size_t)c * Npad;
    unsigned long long a = base[i], b = base[l];
    bool up = ((i & k) != 0);                    // false -> descending block
    bool sw = up ? (a > b) : (a < b);
    if (sw) { base[i] = b; base[l] = a; }
  }
}

// one block per class: gts, then sequential cumsum producing Lovasz gradient dot
__global__ void lovasz_class_k(const unsigned long long* __restrict__ keys, int n, int Npad,
                               float* __restrict__ lossC, float* __restrict__ presC) {
  __shared__ float sh[256];
  __shared__ float scan[256];
  __shared__ float srun;
  const int c = blockIdx.x, tid = threadIdx.x;
  const unsigned long long* kk = keys + (size_t)c * Npad;
  // pass 1: gts = sum of fg (integer-valued, exact in f32 for n <= 2^24)
  float s = 0.f;
  for (int base = 0; base < n; base += 256) {
    int i = base + tid;
    if (i < n) s += (float)(kk[i] & 1ULL);
  }
  sh[tid] = s; __syncthreads();
  for (int q = 128; q > 0; q >>= 1) { if (tid < q) sh[tid] += sh[tid + q]; __syncthreads(); }
  const float gts = sh[0];
  __syncthreads();
  if (tid == 0) srun = 0.f;
  __syncthreads();
  // pass 2: running cumsum of fg, accumulate e_i * (jac_i - jac_{i-1})
  float acc = 0.f;
  for (int base = 0; base < n; base += 256) {
    int i = base + tid;
    float f = 0.f, e = 0.f;
    if (i < n) {
      unsigned long long kv = kk[i];
      f = (float)(kv & 1ULL);
      unsigned ord = (unsigned)(kv >> 32);
      unsigned bits = (ord & 0x80000000u) ? (ord & 0x7fffffffu) : ~ord;
      e = __uint_as_float(bits);
    }
    scan[tid] = f; __syncthreads();
    for (int off = 1; off < 256; off <<= 1) {
      float v = (tid >= off) ? scan[tid - off] : 0.f;
      __syncthreads();
      scan[tid] += v;
      __syncthreads();
    }
    float run = srun;
    float cum = run + scan[tid];
    float chunkTot = run + scan[255];
    __syncthreads();
    if (tid == 0) srun = chunkTot;
    if (i < n) {
      float inter = gts - cum;
      float uni   = gts + (float)(i + 1) - cum;
      float jac   = 1.f - inter / uni;
      float jd;
      if (i == 0) jd = jac;
      else {
        float cump = cum - f;
        float jacp = 1.f - (gts - cump) / (gts + (float)i - cump);
        jd = jac - jacp;
      }
      acc += e * jd;
    }
    __syncthreads();
  }
  sh[tid] = acc; __syncthreads();
  for (int q = 128; q > 0; q >>= 1) { if (tid < q) sh[tid] += sh[tid + q]; __syncthreads(); }
  if (tid == 0) { lossC[c] = sh[0]; presC[c] = (gts > 0.f) ? 1.f : 0.f; }
}

__global__ void finalize_lovasz_k(const float* __restrict__ lossC, const float* __restrict__ presC,
                                  int C, float scale, float* __restrict__ out) {
  if (threadIdx.x == 0) {
    float t = 0.f, p = 0.f;
    for (int c = 0; c < C; ++c) { t += lossC[c] * presC[c]; p += presC[c]; }
    out[0] += scale * t / fmaxf(p, 1.f);
  }
}

// ------------------------------------------------------------------ host side
static void seg_loss_emit(const float* logits, const int* lab, int n, int Npad, float scale,
                          float* loss, float* partial, unsigned long long* keys,
                          float* lossC, float* presC, hipStream_t s) {
  const int C = 20, G = 240;
  ce_partial_k<<<G, 256, 0, s>>>(logits, lab, n, C, partial);
  finalize_scaled_k<<<1, 256, 0, s>>>(partial, G, scale / (float)n, loss);
  build_keys_k<<<(Npad + 255) / 256, 256, 0, s>>>(logits, lab, n, Npad, C, keys);
  const long long tot = (long long)C * Npad;
  const unsigned blocks = (unsigned)((tot + 255) / 256);
  for (int k = 2; k <= Npad; k <<= 1)
    for (int j = k >> 1; j > 0; j >>= 1)
      bitonic_pass_k<<<blocks, 256, 0, s>>>(keys, Npad, C, k, j);
  lovasz_class_k<<<C, 256, 0, s>>>(keys, n, Npad, lossC, presC);
  finalize_lovasz_k<<<1, 32, 0, s>>>(lossC, presC, C, scale, loss);
}

extern "C" void kernel_launch(void* const* d_in, const int* in_sizes, int n_in,
                              void* d_out, int out_size, void* d_ws, size_t ws_size,
                              hipStream_t stream) {
  (void)in_sizes; (void)n_in; (void)out_size; (void)ws_size;
  const float* img_feat = (const float*)d_in[0];
  const float* pts_feat = (const float*)d_in[1];
  const int*   coors    = (const int*)d_in[2];
  const int*   labels   = (const int*)d_in[3];
  const int*   img_lab  = (const int*)d_in[4];
  const int*   p2img    = (const int*)d_in[5];
  const float *w3a=(const float*)d_in[6],  *b3a=(const float*)d_in[7];
  const float *w3b=(const float*)d_in[8],  *b3b=(const float*)d_in[9];
  const float *wfa=(const float*)d_in[10], *bfa=(const float*)d_in[11];
  const float *wfb=(const float*)d_in[12], *bfb=(const float*)d_in[13];
  const float *fc1w=(const float*)d_in[14],*fc1b=(const float*)d_in[15];
  const float *fc2w=(const float*)d_in[16],*fc2b=(const float*)d_in[17];
  const float *fc3w=(const float*)d_in[18],*fc3b=(const float*)d_in[19];
  const float *c1w=(const float*)d_in[20], *c1b=(const float*)d_in[21];
  const float *bn1g=(const float*)d_in[22],*bn1b=(const float*)d_in[23];
  const float *c2w=(const float*)d_in[24], *c2b=(const float*)d_in[25];
  const float *bn2g=(const float*)d_in[26],*bn2b=(const float*)d_in[27];
  const float *clw1=(const float*)d_in[28],*clb1=(const float*)d_in[29];
  const float *clw2=(const float*)d_in[30],*clb2=(const float*)d_in[31];

  const int S = 2, Nv = 80000, N = 240000, Mm = 30000, Nimg = 60000, Np = 120000,
            Hh = 64, Cc = 20;
  const int NPAD_V = 131072, NPAD_I = 65536;

  char* base = (char*)d_ws; size_t off = 0;
  auto carve = [&](size_t bytes) -> void* {
    void* p = base + off; off = (off + bytes + 255) & ~(size_t)255; return p;
  };
  float* feats = (float*)carve(sizeof(float) * (size_t)Nimg * S * Hh);    // [Nimg, S*H]
  float* hbuf  = (float*)carve(sizeof(float) * (size_t)Nv * 128);
  float* pred3d= (float*)carve(sizeof(float) * (size_t)Nv * Cc);
  int*   counts= (int*)  carve(sizeof(int)   * (size_t)Nv * Cc);
  int*   voxlab= (int*)  carve(sizeof(int)   * (size_t)Nv);
  int*   idx   = (int*)  carve(sizeof(int)   * (size_t)Nimg);
  float* pbuf  = (float*)carve(sizeof(float) * (size_t)Nimg * Hh);
  float* ppred = (float*)carve(sizeof(float) * (size_t)Nimg * Cc);
  float* att   = (float*)carve(sizeof(float) * (size_t)Nimg * 32);
  float* attw  = (float*)carve(sizeof(float) * (size_t)Nimg * 2);
  float* y1    = (float*)carve(sizeof(float) * (size_t)Nimg * Hh);
  float* y2    = (float*)carve(sizeof(float) * (size_t)Nimg * Hh);
  float* fpred = (float*)carve(sizeof(float) * (size_t)Nimg * Cc);
  unsigned long long* keys =
      (unsigned long long*)carve(sizeof(unsigned long long) * (size_t)Cc * NPAD_V);
  float* mu1  = (float*)carve(sizeof(float) * 64);
  float* var1 = (float*)carve(sizeof(float) * 64);
  float* mu2  = (float*)carve(sizeof(float) * 64);
  float* var2 = (float*)carve(sizeof(float) * 64);
  float* partial = (float*)carve(sizeof(float) * 256);
  float* lossC = (float*)carve(sizeof(float) * Cc);
  float* presC = (float*)carve(sizeof(float) * Cc);

  float* loss = (float*)d_out;
  zero_f32_k<<<1, 32, 0, stream>>>(loss, 1);

  for (int s = 0; s < S; ++s) {
    const float* pf = pts_feat + (size_t)s * Nv * Hh;
    const float* vf = img_feat + (size_t)s * Nimg * Hh;
    const int*   ci = coors + (size_t)s * N;
    // pred3d = relu(pf@w3a+b3a)@w3b+b3b
    launch_gemm(pf, Hh, w3a + (size_t)s * Hh * 128, 128, b3a + s * 128,
                hbuf, 128, Nv, 128, Hh, ACT_RELU, stream);
    launch_gemm(hbuf, 128, w3b + (size_t)s * 128 * Cc, Cc, b3b + s * Cc,
                pred3d, Cc, Nv, Cc, 128, ACT_NONE, stream);
    // voxel majority vote
    zero_i32_k<<<(Nv * Cc + 255) / 256, 256, 0, stream>>>(counts, Nv * Cc);
    scatter_counts_k<<<(N + 255) / 256, 256, 0, stream>>>(ci, labels, counts, N, Cc);
    vox_argmax_k<<<(Nv + 255) / 256, 256, 0, stream>>>(counts, voxlab, Nv, Cc);
    seg_loss_emit(pred3d, voxlab, Nv, NPAD_V, 1.0f, loss, partial, keys, lossC, presC, stream);
    // gathers to image-projected points
    build_idx_k<<<(Nimg + 255) / 256, 256, 0, stream>>>(ci, p2img, idx, Nimg, Mm, Np);
    gather_rows_k<<<(Nimg * Hh + 255) / 256, 256, 0, stream>>>(pf, idx, pbuf, Nimg, Hh);
    gather_rows_k<<<(Nimg * Cc + 255) / 256, 256, 0, stream>>>(pred3d, idx, ppred, Nimg, Cc);
    // attention weights
    launch_gemm(pbuf, Hh, fc1w + (size_t)s * Hh * 16, 16, fc1b + s * 16,
                att, 32, Nimg, 16, Hh, ACT_NONE, stream);
    launch_gemm(vf, Hh, fc2w + (size_t)s * Hh * 16, 16, fc2b + s * 16,
                att + 16, 32, Nimg, 16, Hh, ACT_NONE, stream);
    launch_gemm(att, 32, fc3w + (size_t)s * 32 * 2, 2, fc3b + s * 2,
                attw, 2, Nimg, 2, 32, ACT_SIG, stream);
    // conv1/conv2 + BN + relu + weighted fuse into feats[:, s*H : s*H+H]
    launch_gemm(pbuf, Hh, c1w + (size_t)s * Hh * Hh, Hh, c1b + s * Hh,
                y1, Hh, Nimg, Hh, Hh, ACT_NONE, stream);
    launch_gemm(vf, Hh, c2w + (size_t)s * Hh * Hh, Hh, c2b + s * Hh,
                y2, Hh, Nimg, Hh, Hh, ACT_NONE, stream);
    bn_stats_k<<<64, 256, 0, stream>>>(y1, Nimg, Hh, mu1, var1);
    bn_stats_k<<<64, 256, 0, stream>>>(y2, Nimg, Hh, mu2, var2);
    fuse_k<<<(Nimg * Hh + 255) / 256, 256, 0, stream>>>(
        y1, y2, mu1, var1, bn1g + s * Hh, bn1b + s * Hh,
        mu2, var2, bn2g + s * Hh, bn2b + s * Hh, attw, feats, Nimg, Hh, S * Hh, s * Hh);
    // fuse classifier
    launch_gemm(feats + s * Hh, S * Hh, wfa + (size_t)s * Hh * 128, 128, bfa + s * 128,
                hbuf, 128, Nimg, 128, Hh, ACT_RELU, stream);
    launch_gemm(hbuf, 128, wfb + (size_t)s * 128 * Cc, Cc, bfb + s * Cc,
                fpred, Cc, Nimg, Cc, 128, ACT_NONE, stream);
    seg_loss_emit(fpred, img_lab, Nimg, NPAD_I, LSEG2D / (float)S,
                  loss, partial, keys, lossC, presC, stream);
    // KL(pts_pred || fuse_pred), mean over all elements
    kl_partial_k<<<240, 256, 0, stream>>>(fpred, ppred, Nimg, Cc, partial);
    finalize_scaled_k<<<1, 256, 0, stream>>>(
        partial, 240, (LXM / (float)S) / (float)((size_t)Nimg * Cc), loss);
  }
  // final classifier over concatenated feats
  launch_gemm(feats, S * Hh, clw1, 128, clb1, hbuf, 128, Nimg, 128, S * Hh, ACT_RELU, stream);
  launch_gemm(hbuf, 128, clw2, Cc, clb2, fpred, Cc, Nimg, Cc, 128, ACT_NONE, stream);
  seg_loss_emit(fpred, img_lab, Nimg, NPAD_I, 1.0f, loss, partial, keys, lossC, presC, stream);
}